// CasperNet_39135742001177
// MI455X (gfx1250) — compile-verified
//
#include <hip/hip_runtime.h>
#include <math.h>

typedef __attribute__((ext_vector_type(2))) float v2f;
typedef __attribute__((ext_vector_type(8))) float v8f;

#define DIN 256   // input dim
#define HN  64    // hidden neurons
#define ONUM 10   // outputs
#define DT  320   // D + H (weight row stride)
#define WRS 68    // padded LDS stride for recurrent weights (conflict-free b64)
#define HST 20    // padded LDS stride for per-wave h/pre tiles (8B/16B-aligned, conflict-free)

__device__ __forceinline__ v8f wmma4(v2f a, v2f b, v8f c) {
  // V_WMMA_F32_16X16X4_F32 : D = A(16x4) * B(4x16) + C(16x16), fp32 exact path
  return __builtin_amdgcn_wmma_f32_16x16x4_f32(false, a, false, b, (short)0, c,
                                               false, false);
}

__device__ __forceinline__ float fast_sigmoid(float s) {
  // v_exp_f32 (+ fused -log2e scale) + v_add_f32 + v_rcp_f32: short dependent
  // chain instead of the IEEE v_div_scale/v_div_fmas sequence.
  return __builtin_amdgcn_rcpf(1.0f + __expf(-s));
}

// ---------------------------------------------------------------------------
// Prep: pack B-matrix fragments for the main K-loop into d_ws in the exact
// per-lane VGPR layout of V_WMMA (lane l<16 holds K=k..k+1 for N=l,
// lanes 16-31 hold K=k+2..k+3).  5 N-tiles: 4 hidden (W_h) + 1 output (W_out).
// Layout: ws[(t*5 + nt)*32 + lane] as float2 -> 256B fully-coalesced per wave.
// ---------------------------------------------------------------------------
__global__ __launch_bounds__(256) void casper_prep(const float* __restrict__ Wh,
                                                   const float* __restrict__ Wout,
                                                   float* __restrict__ wsB) {
  int tid = blockIdx.x * blockDim.x + threadIdx.x;
  if (tid >= 64 * 5 * 32) return;
  int t  = tid / 160;         // K-step (K = 4*t)
  int nt = (tid % 160) / 32;  // N-tile
  int l  = tid & 31;          // lane
  int kb = 4 * t + ((l >> 4) << 1);
  float v0 = 0.f, v1 = 0.f;
  if (nt < 4) {
    int n = nt * 16 + (l & 15);
    v0 = Wh[n * DT + kb];
    v1 = Wh[n * DT + kb + 1];
  } else {
    int o = l & 15;
    if (o < ONUM) {
      v0 = Wout[o * DT + kb];
      v1 = Wout[o * DT + kb + 1];
    }
  }
  ((v2f*)wsB)[tid] = (v2f){v0, v1};
}

// ---------------------------------------------------------------------------
// Main: 8 waves/block, each wave owns a 16-row tile.
//   1) pre[16x64] and out_in[16x16] via f32 WMMA over K=256 (5 tiles/K-step)
//   2) cascade in 4 neuron-blocks of 16: cross-block updates via WMMA
//      (h_block @ Wr^T), within-block 16 serial sigmoid steps per lane
//   3) out += h[16x64] @ Wor^T via WMMA, add biases, store
// ---------------------------------------------------------------------------
__global__ __launch_bounds__(256) void casper_main(
    const float* __restrict__ x, const float* __restrict__ Wh,
    const float* __restrict__ bh, const float* __restrict__ Wout,
    const float* __restrict__ bo, const float* __restrict__ wsB,
    float* __restrict__ out, int nrows) {
  __shared__ __align__(16) float WrL[HN * WRS];   // W_h[:,256:320], padded
  __shared__ __align__(16) float WorL[16 * WRS];  // W_out[:,256:320], padded
  __shared__ float bhL[HN];
  __shared__ float boL[16];
  __shared__ __align__(16) float hbuf[8][4][16 * HST];  // per-wave h/pre tiles

  const int tid  = threadIdx.x;
  const int lane = tid & 31;
  const int wid  = tid >> 5;
  const int mrow  = lane & 15;   // M (row) index / N (col) index of fragments
  const int khalf = lane >> 4;   // which K-pair half this lane holds

  // ---- cooperative LDS fill (small, coalesced) ----
  for (int idx = tid; idx < HN * WRS; idx += 256) {
    int i = idx / WRS, j = idx % WRS;
    WrL[idx] = (j < HN) ? Wh[i * DT + DIN + j] : 0.f;
  }
  for (int idx = tid; idx < 16 * WRS; idx += 256) {
    int o = idx / WRS, j = idx % WRS;
    WorL[idx] = (o < ONUM && j < HN) ? Wout[o * DT + DIN + j] : 0.f;
  }
  if (tid < HN) bhL[tid] = bh[tid];
  if (tid < 16) boL[tid] = (tid < ONUM) ? bo[tid] : 0.f;
  __syncthreads();

  const int row0 = blockIdx.x * 128 + wid * 16;
  const float* xl = x + (size_t)(row0 + mrow) * DIN + khalf * 2;

  v8f cP[4];
  v8f cOut = {0.f, 0.f, 0.f, 0.f, 0.f, 0.f, 0.f, 0.f};
#pragma unroll
  for (int g = 0; g < 4; ++g) cP[g] = cOut;

  // ---- main GEMM over K=256: x-tile vs [W_h_in | W_out_in] ----
  const v2f* Bws = (const v2f*)wsB;
#pragma unroll 2
  for (int t = 0; t < 64; ++t) {
    v2f a  = *(const v2f*)(xl + 4 * t);  // A: 16x4 fp32 fragment (global)
    v2f b0 = Bws[(t * 5 + 0) * 32 + lane];
    v2f b1 = Bws[(t * 5 + 1) * 32 + lane];
    v2f b2 = Bws[(t * 5 + 2) * 32 + lane];
    v2f b3 = Bws[(t * 5 + 3) * 32 + lane];
    v2f b4 = Bws[(t * 5 + 4) * 32 + lane];
    cP[0] = wmma4(a, b0, cP[0]);
    cP[1] = wmma4(a, b1, cP[1]);
    cP[2] = wmma4(a, b2, cP[2]);
    cP[3] = wmma4(a, b3, cP[3]);
    cOut  = wmma4(a, b4, cOut);
  }

  // ---- cascade over 4 blocks of 16 neurons ----
#pragma unroll
  for (int g = 0; g < 4; ++g) {
    v8f acc = cP[g];
    // cross-block triangular contributions: acc += h[g'] @ Wr[g, g']^T
#pragma unroll
    for (int gp = 0; gp < 4; ++gp) {
      if (gp < g) {
#pragma unroll
        for (int kk = 0; kk < 4; ++kk) {
          int jj = 4 * kk + khalf * 2;
          v2f a = *(const v2f*)&hbuf[wid][gp][mrow * HST + jj];
          v2f b = *(const v2f*)&WrL[(g * 16 + mrow) * WRS + gp * 16 + jj];
          acc = wmma4(a, b, acc);
        }
      }
    }
    // dump C fragment (conflict-free column writes)
    {
      int mb = khalf * 8;
#pragma unroll
      for (int r = 0; r < 8; ++r)
        hbuf[wid][g][(mb + r) * HST + mrow] = acc[r];
    }
    __syncthreads();
    // within-block serial recurrence: lane = row (upper lanes duplicate)
    {
      float hl[16];
#pragma unroll
      for (int i = 0; i < 16; ++i) {
        float s = hbuf[wid][g][mrow * HST + i] + bhL[g * 16 + i];
#pragma unroll
        for (int j = 0; j < i; ++j)
          s += WrL[(g * 16 + i) * WRS + g * 16 + j] * hl[j];  // LDS broadcast
        float h = fast_sigmoid(s);
        hl[i] = h;
        hbuf[wid][g][mrow * HST + i] = h;  // overwrite pre with h
      }
    }
    __syncthreads();
  }

  // ---- output hidden part: cOut += h[16x64] @ Wor^T ----
#pragma unroll
  for (int g = 0; g < 4; ++g) {
#pragma unroll
    for (int kk = 0; kk < 4; ++kk) {
      int jj = 4 * kk + khalf * 2;
      v2f a = *(const v2f*)&hbuf[wid][g][mrow * HST + jj];
      v2f b = *(const v2f*)&WorL[mrow * WRS + g * 16 + jj];
      cOut = wmma4(a, b, cOut);
    }
  }

  // ---- store (fragment column = output index) ----
  if (mrow < ONUM) {
    int mb = khalf * 8;
    float bb = boL[mrow];
#pragma unroll
    for (int r = 0; r < 8; ++r)
      out[(size_t)(row0 + mb + r) * ONUM + mrow] = cOut[r] + bb;
  }
}

extern "C" void kernel_launch(void* const* d_in, const int* in_sizes, int n_in,
                              void* d_out, int out_size, void* d_ws,
                              size_t ws_size, hipStream_t stream) {
  const float* x   = (const float*)d_in[0];
  const float* Wh  = (const float*)d_in[1];
  const float* bhp = (const float*)d_in[2];
  const float* Wo  = (const float*)d_in[3];
  const float* bop = (const float*)d_in[4];
  float* outp = (float*)d_out;
  float* wsB  = (float*)d_ws;  // needs 80 KB of scratch

  int nrows   = in_sizes[0] / DIN;   // 131072
  int nblocks = nrows / 128;         // 1024 blocks x 256 threads (8 waves)

  casper_prep<<<40, 256, 0, stream>>>(Wh, Wo, wsB);
  casper_main<<<nblocks, 256, 0, stream>>>(x, Wh, bhp, Wo, bop, wsB, outp,
                                           nrows);
}